// HGNNConv_46342697124074
// MI455X (gfx1250) — compile-verified
//
#include <hip/hip_runtime.h>
#include <math.h>

typedef __attribute__((ext_vector_type(2))) float v2f;
typedef __attribute__((ext_vector_type(8))) float v8f;

#define N_NODES 50000
#define N_EDGES 25000
#define NNZ     800000
#define IN_DIM  256
#define OUT_DIM 256

#define LDS_STRIDE 260   // 256 + 4 pad: row stride 260 dwords -> banks advance by 4 per row

// ---------------------------------------------------------------------------
// zero a float buffer
// ---------------------------------------------------------------------------
__global__ void k_zero(float* __restrict__ p, int n) {
    int i = blockIdx.x * blockDim.x + threadIdx.x;
    if (i < n) p[i] = 0.0f;
}

// ---------------------------------------------------------------------------
// degree histograms: deg_v[node_idx[e]] += 1 ; deg_e[edge_idx[e]] += 1
// (float atomic adds of 1.0 are exact for counts << 2^24)
// ---------------------------------------------------------------------------
__global__ void k_deg(const int* __restrict__ ni, const int* __restrict__ ei,
                      float* __restrict__ degv, float* __restrict__ dege, int nnz) {
    int i = blockIdx.x * blockDim.x + threadIdx.x;
    if (i < nnz) {
        atomicAdd(&degv[ni[i]], 1.0f);
        atomicAdd(&dege[ei[i]], 1.0f);
    }
}

// ---------------------------------------------------------------------------
// d_v = deg^-0.5 (0 if deg==0), d_e = deg^-1 (0 if deg==0), in place
// ---------------------------------------------------------------------------
__global__ void k_inv(float* __restrict__ degv, float* __restrict__ dege) {
    int i = blockIdx.x * blockDim.x + threadIdx.x;
    if (i < N_NODES) {
        float d = degv[i];
        degv[i] = (d > 0.0f) ? (1.0f / sqrtf(d)) : 0.0f;
    }
    if (i < N_EDGES) {
        float d = dege[i];
        dege[i] = (d > 0.0f) ? (1.0f / d) : 0.0f;
    }
}

// ---------------------------------------------------------------------------
// Z = d_v[:,None] * (X @ W + b)   via V_WMMA_F32_16X16X4_F32
// block = 256 threads (8 waves). Block owns a 16(M) x 256(N) strip.
// Wave w owns N-cols [32w, 32w+32) as two 16x16 accumulators.
// X strip staged in LDS (padded); W/b read from global (hot in L2/WGP$).
// ---------------------------------------------------------------------------
__global__ void __launch_bounds__(256) k_gemm(const float* __restrict__ X,
                                              const float* __restrict__ W,
                                              const float* __restrict__ bias,
                                              const float* __restrict__ d_v,
                                              float* __restrict__ Z) {
    __shared__ float sA[16 * LDS_STRIDE];

    const int tid   = threadIdx.x;
    const int wave  = tid >> 5;
    const int lane  = tid & 31;
    const int half  = lane >> 4;      // 0: lanes 0-15, 1: lanes 16-31
    const int l16   = lane & 15;
    const int mBase = blockIdx.x * 16;

    // stage X[mBase .. mBase+16, 0..256) -> LDS (coalesced b32 loads)
    for (int i = tid; i < 16 * 256; i += 256) {
        int r = i >> 8;
        int c = i & 255;
        sA[r * LDS_STRIDE + c] = X[(size_t)(mBase + r) * IN_DIM + c];
    }
    __syncthreads();

    const int nBase = wave * 32;

    v8f acc0 = {};
    v8f acc1 = {};

    for (int kk = 0; kk < IN_DIM; kk += 4) {
        const int ka = kk + 2 * half;  // A: lanes 0-15 hold K=kk,kk+1; lanes 16-31 hold K=kk+2,kk+3
        v2f a;
        a.x = sA[l16 * LDS_STRIDE + ka];
        a.y = sA[l16 * LDS_STRIDE + ka + 1];

        // B mirrors A across the N dimension: lane group half picks K rows ka, ka+1
        const float* w0 = W + (size_t)ka * OUT_DIM;
        v2f b0, b1;
        b0.x = w0[nBase + l16];
        b0.y = w0[OUT_DIM + nBase + l16];
        b1.x = w0[nBase + 16 + l16];
        b1.y = w0[OUT_DIM + nBase + 16 + l16];

        acc0 = __builtin_amdgcn_wmma_f32_16x16x4_f32(false, a, false, b0,
                                                     (short)0, acc0, false, false);
        acc1 = __builtin_amdgcn_wmma_f32_16x16x4_f32(false, a, false, b1,
                                                     (short)0, acc1, false, false);
    }

    const int col0 = nBase + l16;
    const float bc0 = bias[col0];
    const float bc1 = bias[col0 + 16];

    // C/D layout: VGPR r holds M = r (lanes 0-15) / M = r+8 (lanes 16-31), N = l16
#pragma unroll
    for (int r = 0; r < 8; ++r) {
        const int row   = mBase + r + 8 * half;
        const float s   = d_v[row];
        float* zr       = Z + (size_t)row * OUT_DIM;
        zr[col0]        = (acc0[r] + bc0) * s;
        zr[col0 + 16]   = (acc1[r] + bc1) * s;
    }
}

// ---------------------------------------------------------------------------
// step1[edge_idx[e], :] += Z[node_idx[e], :]
// 64 threads per incidence entry, float4 per thread, non-returning f32 atomics
// ---------------------------------------------------------------------------
__global__ void k_scatter1(const float* __restrict__ Z,
                           const int* __restrict__ ni, const int* __restrict__ ei,
                           float* __restrict__ step1) {
    int t = blockIdx.x * blockDim.x + threadIdx.x;
    int e = t >> 6;
    int c = (t & 63) << 2;
    if (e < NNZ) {
        int n = ni[e];
        int m = ei[e];
        const float4 z = *(const float4*)(Z + (size_t)n * OUT_DIM + c);
        float* dst = step1 + (size_t)m * OUT_DIM + c;
        atomicAdd(dst + 0, z.x);
        atomicAdd(dst + 1, z.y);
        atomicAdd(dst + 2, z.z);
        atomicAdd(dst + 3, z.w);
    }
}

// ---------------------------------------------------------------------------
// out[node_idx[e], :] += d_e[edge_idx[e]] * step1[edge_idx[e], :]
// (D_e scaling fused into the gather)
// ---------------------------------------------------------------------------
__global__ void k_scatter2(const float* __restrict__ step1,
                           const float* __restrict__ d_e,
                           const int* __restrict__ ni, const int* __restrict__ ei,
                           float* __restrict__ out) {
    int t = blockIdx.x * blockDim.x + threadIdx.x;
    int e = t >> 6;
    int c = (t & 63) << 2;
    if (e < NNZ) {
        int n = ni[e];
        int m = ei[e];
        float s = d_e[m];
        const float4 v = *(const float4*)(step1 + (size_t)m * OUT_DIM + c);
        float* dst = out + (size_t)n * OUT_DIM + c;
        atomicAdd(dst + 0, s * v.x);
        atomicAdd(dst + 1, s * v.y);
        atomicAdd(dst + 2, s * v.z);
        atomicAdd(dst + 3, s * v.w);
    }
}

// ---------------------------------------------------------------------------
// out[n, :] *= d_v[n]   (float4 per thread; 4 | 256 so all 4 share one row)
// ---------------------------------------------------------------------------
__global__ void k_finalize(float* __restrict__ out, const float* __restrict__ d_v) {
    int t = blockIdx.x * blockDim.x + threadIdx.x;
    const int total4 = N_NODES * OUT_DIM / 4;
    if (t < total4) {
        int i = t << 2;
        float s = d_v[i >> 8];
        float4* p = (float4*)out + t;
        float4 v = *p;
        v.x *= s; v.y *= s; v.z *= s; v.w *= s;
        *p = v;
    }
}

// ---------------------------------------------------------------------------
// launch
// ---------------------------------------------------------------------------
extern "C" void kernel_launch(void* const* d_in, const int* in_sizes, int n_in,
                              void* d_out, int out_size, void* d_ws, size_t ws_size,
                              hipStream_t stream) {
    const float* X    = (const float*)d_in[0];
    const int*   ni   = (const int*)d_in[1];
    const int*   ei   = (const int*)d_in[2];
    const float* W    = (const float*)d_in[3];
    const float* bias = (const float*)d_in[4];
    float*       out  = (float*)d_out;

    // workspace layout (bytes)
    char* ws = (char*)d_ws;
    float* Z     = (float*)(ws);                              // 50000*256*4 = 51,200,000
    float* step1 = (float*)(ws + 51200000);                   // 25000*256*4 = 25,600,000
    float* dv    = (float*)(ws + 51200000 + 25600000);        // 50000*4     =    200,000
    float* de    = (float*)(ws + 51200000 + 25600000 + 200000); // 25000*4   =    100,000

    const int T = 256;

    // 1) zero accumulators: step1, degrees (dv+de contiguous), d_out
    {
        int n1 = N_EDGES * OUT_DIM;                 // 6,400,000
        k_zero<<<(n1 + T - 1) / T, T, 0, stream>>>(step1, n1);
        int n2 = N_NODES + N_EDGES;                 // 75,000 (dv,de contiguous)
        k_zero<<<(n2 + T - 1) / T, T, 0, stream>>>(dv, n2);
        int n3 = N_NODES * OUT_DIM;                 // 12,800,000
        k_zero<<<(n3 + T - 1) / T, T, 0, stream>>>(out, n3);
    }

    // 2) degree histograms
    k_deg<<<(NNZ + T - 1) / T, T, 0, stream>>>(ni, ei, dv, de, NNZ);

    // 3) invert degrees in place -> d_v, d_e
    k_inv<<<(N_NODES + T - 1) / T, T, 0, stream>>>(dv, de);

    // 4) Z = d_v * (X @ W + b)  (WMMA f32 16x16x4)
    k_gemm<<<N_NODES / 16, T, 0, stream>>>(X, W, bias, dv, Z);

    // 5) step1 = H^T @ Z (scatter-add into edges)
    {
        long long threads = (long long)NNZ * 64;
        k_scatter1<<<(unsigned)((threads + T - 1) / T), T, 0, stream>>>(Z, ni, ei, step1);
    }

    // 6) out += H @ (D_e * step1) (scatter-add into nodes, d_e fused)
    {
        long long threads = (long long)NNZ * 64;
        k_scatter2<<<(unsigned)((threads + T - 1) / T), T, 0, stream>>>(step1, de, ni, ei, out);
    }

    // 7) out *= d_v
    {
        int total4 = N_NODES * OUT_DIM / 4;
        k_finalize<<<(total4 + T - 1) / T, T, 0, stream>>>(out, dv);
    }
}